// MC_DataPMLayer_83854941487175
// MI455X (gfx1250) — compile-verified
//
#include <hip/hip_runtime.h>

// Problem constants (match reference)
#define NDIM 320
#define HWSZ (320 * 320)
#define NBATCH 4
#define NCOIL 12
#define NSLICE (NBATCH * NCOIL)   // 48 independent complex images
#define CG_ITERS 10
#define CG_TOL 1e-10f

// GEMM tiling
#define BM 64          // block tile M (4 waves x 16 rows)
#define BN 64          // block tile N
#define BK 32          // K depth staged in LDS per step
#define LDS_STRIDE 80  // 64 data + 16 pad DWORDs per row (bank-conflict-free)

typedef __attribute__((ext_vector_type(2))) float v2f;
typedef __attribute__((ext_vector_type(8))) float v8f;
typedef unsigned int u32x4 __attribute__((ext_vector_type(4)));
typedef int i32x4 __attribute__((ext_vector_type(4)));
typedef int i32x8 __attribute__((ext_vector_type(8)));

// ---------------------------------------------------------------------------
// TDM: issue a 2D tile load (tileH rows x tileW f32) from global (row stride
// strideElems f32) into LDS at byte offset ldsOff, padding each 64-DWORD row
// by 16 DWORDs (LDS row stride = 80 DWORDs). D# built per CDNA5 ISA 8.3-8.5.
// This toolchain exposes the 6-arg builtin:
//   (u32x4 g0, i32x8 g1, i32x4 g2, i32x4 g3, i32x8 extra, i32 cpol)
// ---------------------------------------------------------------------------
__device__ __forceinline__ void tdm_load_tile_2d(const float* gsrc, unsigned ldsOff,
                                                 int tileW, int tileH, int strideElems) {
    const unsigned long long ga = (unsigned long long)(uintptr_t)gsrc;
    u32x4 g0;
    g0[0] = 1u;                                   // count=1, user descriptor
    g0[1] = ldsOff;                               // lds_addr (bytes)
    g0[2] = (unsigned)(ga & 0xFFFFFFFFull);       // global_addr[31:0]
    g0[3] = (unsigned)((ga >> 32) & 0x01FFFFFFull) | 0x80000000u; // addr[56:32] | type=2

    const int D0 = 1 << 20;                       // tensor_dim0 (never clips our tiles)
    const int D1 = 1 << 20;                       // tensor_dim1
    i32x8 g1;
    g1[0] = (2 << 16)                             // data_size = 4 bytes
          | (1 << 20)                             // pad_enable
          | (5 << 22)                             // pad_interval: 2^(5+1)=64 DWORDs
          | (15 << 25);                           // pad_amount: 15+1=16 DWORDs
    g1[1] = (D0 & 0xFFFF) << 16;                  // dim0[15:0] (atomic_barrier_addr=0)
    g1[2] = ((D0 >> 16) & 0xFFFF) | ((D1 & 0xFFFF) << 16);  // dim0[31:16] | dim1[15:0]
    g1[3] = ((D1 >> 16) & 0xFFFF) | (tileW << 16);          // dim1[31:16] | tile_dim0
    g1[4] = tileH & 0xFFFF;                       // tile_dim1 (tile_dim2 = 0)
    g1[5] = strideElems;                          // tensor_dim0_stride[31:0]
    g1[6] = 0;                                    // stride[47:32] | dim1_stride[15:0]
    g1[7] = 0;

    i32x4 gz4 = {0, 0, 0, 0};                     // groups 2/3 unused for 2D tiles
    i32x8 gz8 = {0, 0, 0, 0, 0, 0, 0, 0};
    __builtin_amdgcn_tensor_load_to_lds(g0, g1, gz4, gz4, gz8, 0);
}

// ---------------------------------------------------------------------------
// Build centered ortho DFT matrix: W[k,n] = exp(-2*pi*i*(k-160)*(n-160)/320)/sqrt(320)
// ---------------------------------------------------------------------------
__global__ __launch_bounds__(256)
void build_w_kernel(float* __restrict__ Wr, float* __restrict__ Wi) {
    const int idx = blockIdx.x * 256 + threadIdx.x;
    if (idx >= HWSZ) return;
    const int k = idx / NDIM;
    const int n = idx % NDIM;
    long p = (long)(k - NDIM / 2) * (long)(n - NDIM / 2);
    int r = (int)(p % NDIM);
    if (r < 0) r += NDIM;
    const float theta = -2.0f * 3.14159265358979323846f * (float)r / (float)NDIM;
    const float scale = 0.05590169943749474f; // 1/sqrt(320)
    float s, c;
    __sincosf(theta, &s, &c);
    Wr[idx] = scale * c;
    Wi[idx] = scale * s;
}

// ---------------------------------------------------------------------------
// Init: rhs = x0 + lam*z_k ; r = p = rhs ; x = 0 ; done = 0
// ---------------------------------------------------------------------------
__global__ __launch_bounds__(256)
void init_kernel(const float* __restrict__ zk, const float* __restrict__ x0,
                 const float* __restrict__ lamPtr,
                 float* __restrict__ xr, float* __restrict__ xi,
                 float* __restrict__ rr, float* __restrict__ ri,
                 float* __restrict__ pr, float* __restrict__ pi,
                 int* __restrict__ done) {
    const int s = blockIdx.y;
    const long i = (long)blockIdx.x * 256 + threadIdx.x;
    const int b = s / NCOIL, c = s % NCOIL;
    const float lam = lamPtr[0];
    const long reIdx = (((long)b * 2 + 0) * NCOIL + c) * HWSZ + i;
    const long imIdx = (((long)b * 2 + 1) * NCOIL + c) * HWSZ + i;
    const float vr = x0[reIdx] + lam * zk[reIdx];
    const float vi = x0[imIdx] + lam * zk[imIdx];
    const long o = (long)s * HWSZ + i;
    rr[o] = vr; ri[o] = vi;
    pr[o] = vr; pi[o] = vi;
    xr[o] = 0.0f; xi[o] = 0.0f;
    if (s == 0 && i == 0) *done = 0;
}

// ---------------------------------------------------------------------------
// Batched complex GEMM via V_WMMA_F32_16X16X4_F32 with TDM-staged B tiles.
// C = (conjA? conj(A) : A) x (conjB? conj(B) : B), 320x320 per slice.
// Block: 128 threads (4 waves) -> 64x64 complex C tile.
// Wave w computes rows [16w,16w+16) x 64 cols (4 subtiles), A frag reused 4x.
// B tile (BK x BN, re & im) staged in LDS by the Tensor Data Mover.
// ---------------------------------------------------------------------------
__global__ __launch_bounds__(128)
void cgemm_wmma_kernel(const float* __restrict__ Ar, const float* __restrict__ Ai, long sA,
                       const float* __restrict__ Br, const float* __restrict__ Bi, long sB,
                       float* __restrict__ Cr, float* __restrict__ Ci,
                       int conjA, int conjB,
                       const float* __restrict__ mask,        // nullable
                       const float* __restrict__ addR,        // nullable
                       const float* __restrict__ addI,        // nullable
                       const float* __restrict__ lamPtr) {
    __shared__ float BsR[BK * LDS_STRIDE];
    __shared__ float BsI[BK * LDS_STRIDE];

    const int s = blockIdx.z;
    const int rowBase = blockIdx.y * BM;
    const int colBase = blockIdx.x * BN;
    const int tid = threadIdx.x;
    const int wave = tid >> 5;
    const int lane = tid & 31;
    const int half = lane >> 4;   // 0 or 1
    const int l16 = lane & 15;
    const int kq = half * 2;
    const float sgnA = conjA ? -1.0f : 1.0f;
    const float sgnB = conjB ? -1.0f : 1.0f;

    const float* ar = Ar + (long)s * sA;
    const float* ai = Ai + (long)s * sA;
    const float* br = Br + (long)s * sB;
    const float* bi = Bi + (long)s * sB;

    v8f accR[4] = {};   // 4 column subtiles, real part
    v8f accI[4] = {};   // imag part

    const int aRow = rowBase + wave * 16 + l16;   // A frag: M = this lane's row
    const unsigned bsrOff = (unsigned)(uintptr_t)&BsR[0];
    const unsigned bsiOff = (unsigned)(uintptr_t)&BsI[0];

    for (int kt = 0; kt < NDIM; kt += BK) {
        // Stage B tile [kt..kt+BK) x [colBase..colBase+BN) into LDS via TDM.
        if (tid < 32) {
            tdm_load_tile_2d(br + (long)kt * NDIM + colBase, bsrOff, BN, BK, NDIM);
            tdm_load_tile_2d(bi + (long)kt * NDIM + colBase, bsiOff, BN, BK, NDIM);
            __builtin_amdgcn_s_wait_tensorcnt(0);
        }
        __syncthreads();

#pragma unroll
        for (int kk = 0; kk < BK; kk += 4) {
            // A fragment (16x4 f32): lane row aRow, K = kt+kk+kq+{0,1}
            const float2 arv = *(const float2*)(ar + (long)aRow * NDIM + (kt + kk + kq));
            const float2 aiv = *(const float2*)(ai + (long)aRow * NDIM + (kt + kk + kq));
            v2f fAr  = { arv.x, arv.y };
            v2f fAi  = { sgnA * aiv.x, sgnA * aiv.y };
            v2f fAni = { -fAi.x, -fAi.y };

#pragma unroll
            for (int t = 0; t < 4; ++t) {
                // B fragment (4x16): col = t*16 + l16, K rows kk+kq+{0,1} from LDS
                const int r0 = (kk + kq) * LDS_STRIDE + t * 16 + l16;
                v2f fBr = { BsR[r0], BsR[r0 + LDS_STRIDE] };
                v2f fBi = { sgnB * BsI[r0], sgnB * BsI[r0 + LDS_STRIDE] };

                accR[t] = __builtin_amdgcn_wmma_f32_16x16x4_f32(false, fAr,  false, fBr, (short)0, accR[t], false, false);
                accR[t] = __builtin_amdgcn_wmma_f32_16x16x4_f32(false, fAni, false, fBi, (short)0, accR[t], false, false);
                accI[t] = __builtin_amdgcn_wmma_f32_16x16x4_f32(false, fAr,  false, fBi, (short)0, accI[t], false, false);
                accI[t] = __builtin_amdgcn_wmma_f32_16x16x4_f32(false, fAi,  false, fBr, (short)0, accI[t], false, false);
            }
        }
        __syncthreads();   // all waves done reading before next TDM overwrite
    }

    const float lam = lamPtr[0];
    const int b = s / NCOIL;
    float* cr = Cr + (long)s * HWSZ;
    float* ci = Ci + (long)s * HWSZ;

    // C store: subtile t, VGPR j -> row = rowBase+16*wave+j+8*half, col = colBase+16t+l16
#pragma unroll
    for (int t = 0; t < 4; ++t) {
#pragma unroll
        for (int j = 0; j < 8; ++j) {
            const int row = rowBase + wave * 16 + j + half * 8;
            const int col = colBase + t * 16 + l16;
            const long idx = (long)row * NDIM + col;
            float vr = accR[t][j];
            float vi = accI[t][j];
            if (mask != nullptr) {
                const float mv = mask[(long)b * HWSZ + idx];
                vr *= mv; vi *= mv;
            }
            if (addR != nullptr) {
                vr += lam * addR[(long)s * HWSZ + idx];
                vi += lam * addI[(long)s * HWSZ + idx];
            }
            cr[idx] = vr;
            ci[idx] = vi;
        }
    }
}

// ---------------------------------------------------------------------------
// Per-slice dot: out[s] = sum(ar*br + ai*bi). One block per slice.
// ---------------------------------------------------------------------------
__global__ __launch_bounds__(256)
void dot_kernel(const float* __restrict__ ar, const float* __restrict__ ai,
                const float* __restrict__ br, const float* __restrict__ bi,
                float* __restrict__ out) {
    __shared__ float sbuf[256];
    const int s = blockIdx.x;
    const long base = (long)s * HWSZ;
    float acc = 0.0f;
    for (int i = threadIdx.x; i < HWSZ; i += 256)
        acc += ar[base + i] * br[base + i] + ai[base + i] * bi[base + i];
    sbuf[threadIdx.x] = acc;
    __syncthreads();
    for (int off = 128; off > 0; off >>= 1) {
        if (threadIdx.x < off) sbuf[threadIdx.x] += sbuf[threadIdx.x + off];
        __syncthreads();
    }
    if (threadIdx.x == 0) out[s] = sbuf[0];
}

// done |= !all(rTr > tol)   (sticky, checked at top of each iteration)
__global__ void check_kernel(const float* __restrict__ rTr, int* __restrict__ done) {
    if (threadIdx.x == 0 && blockIdx.x == 0) {
        if (*done == 0) {
            for (int s = 0; s < NSLICE; ++s)
                if (!(rTr[s] > CG_TOL)) { *done = 1; break; }
        }
    }
}

// x += alpha*p ; r -= alpha*Ap   (alpha per slice), frozen when done
__global__ __launch_bounds__(256)
void update_xr_kernel(float* __restrict__ xr, float* __restrict__ xi,
                      float* __restrict__ rr, float* __restrict__ ri,
                      const float* __restrict__ pr, const float* __restrict__ pi,
                      const float* __restrict__ apr, const float* __restrict__ api,
                      const float* __restrict__ rTr, const float* __restrict__ pAp,
                      const int* __restrict__ done) {
    if (*done) return;
    const int s = blockIdx.y;
    const long i = (long)s * HWSZ + (long)blockIdx.x * 256 + threadIdx.x;
    const float alpha = rTr[s] / pAp[s];
    xr[i] += alpha * pr[i];
    xi[i] += alpha * pi[i];
    rr[i] -= alpha * apr[i];
    ri[i] -= alpha * api[i];
}

// p = r + beta*p  (beta = rTrNew/rTrOld per slice), frozen when done
__global__ __launch_bounds__(256)
void update_p_kernel(float* __restrict__ pr, float* __restrict__ pi,
                     const float* __restrict__ rr, const float* __restrict__ ri,
                     const float* __restrict__ rTrN, const float* __restrict__ rTr,
                     const int* __restrict__ done) {
    if (*done) return;
    const int s = blockIdx.y;
    const long i = (long)s * HWSZ + (long)blockIdx.x * 256 + threadIdx.x;
    const float beta = rTrN[s] / rTr[s];
    pr[i] = rr[i] + beta * pr[i];
    pi[i] = ri[i] + beta * pi[i];
}

__global__ void commit_rtr_kernel(float* __restrict__ rTr, const float* __restrict__ rTrN,
                                  const int* __restrict__ done) {
    const int s = threadIdx.x;
    if (s < NSLICE && *done == 0) rTr[s] = rTrN[s];
}

// out (B,2,NC,H,W): [:,0]=Re(x), [:,1]=Im(x)
__global__ __launch_bounds__(256)
void writeback_kernel(const float* __restrict__ xr, const float* __restrict__ xi,
                      float* __restrict__ out) {
    const int s = blockIdx.y;
    const long i = (long)blockIdx.x * 256 + threadIdx.x;
    const int b = s / NCOIL, c = s % NCOIL;
    out[(((long)b * 2 + 0) * NCOIL + c) * HWSZ + i] = xr[(long)s * HWSZ + i];
    out[(((long)b * 2 + 1) * NCOIL + c) * HWSZ + i] = xi[(long)s * HWSZ + i];
}

// ---------------------------------------------------------------------------
extern "C" void kernel_launch(void* const* d_in, const int* in_sizes, int n_in,
                              void* d_out, int out_size, void* d_ws, size_t ws_size,
                              hipStream_t stream) {
    const float* zk   = (const float*)d_in[0];
    const float* x0   = (const float*)d_in[1];
    const float* mask = (const float*)d_in[2];
    const float* lam  = (const float*)d_in[3];
    float* out = (float*)d_out;

    // Workspace layout (floats)
    float* ws = (float*)d_ws;
    float* Wr  = ws;            float* Wi  = Wr  + HWSZ;
    float* xr  = Wi  + HWSZ;    float* xi  = xr  + (long)NSLICE * HWSZ;
    float* rr  = xi  + (long)NSLICE * HWSZ;  float* ri  = rr + (long)NSLICE * HWSZ;
    float* pr  = ri  + (long)NSLICE * HWSZ;  float* pi  = pr + (long)NSLICE * HWSZ;
    float* apr = pi  + (long)NSLICE * HWSZ;  float* api = apr + (long)NSLICE * HWSZ;
    float* tr  = api + (long)NSLICE * HWSZ;  float* ti  = tr + (long)NSLICE * HWSZ;
    float* rTr  = ti   + (long)NSLICE * HWSZ;
    float* pAp  = rTr  + NSLICE;
    float* rTrN = pAp  + NSLICE;
    int*   done = (int*)(rTrN + NSLICE);

    const dim3 gemmGrid(NDIM / BN, NDIM / BM, NSLICE);
    const dim3 gemmBlk(128, 1, 1);
    const dim3 elemGrid(HWSZ / 256, NSLICE, 1);
    const dim3 elemBlk(256, 1, 1);

    build_w_kernel<<<(HWSZ + 255) / 256, 256, 0, stream>>>(Wr, Wi);
    init_kernel<<<elemGrid, elemBlk, 0, stream>>>(zk, x0, lam, xr, xi, rr, ri, pr, pi, done);
    dot_kernel<<<NSLICE, 256, 0, stream>>>(rr, ri, rr, ri, rTr);

    for (int it = 0; it < CG_ITERS; ++it) {
        check_kernel<<<1, 32, 0, stream>>>(rTr, done);

        // Ap = conj(W) * ( (W * p * W) .* mask ) * conj(W) + lam * p
        // t = W * p
        cgemm_wmma_kernel<<<gemmGrid, gemmBlk, 0, stream>>>(
            Wr, Wi, 0L, pr, pi, (long)HWSZ, tr, ti, 0, 0,
            nullptr, nullptr, nullptr, lam);
        // ap = (t * W) .* mask
        cgemm_wmma_kernel<<<gemmGrid, gemmBlk, 0, stream>>>(
            tr, ti, (long)HWSZ, Wr, Wi, 0L, apr, api, 0, 0,
            mask, nullptr, nullptr, lam);
        // t = conj(W) * ap
        cgemm_wmma_kernel<<<gemmGrid, gemmBlk, 0, stream>>>(
            Wr, Wi, 0L, apr, api, (long)HWSZ, tr, ti, 1, 0,
            nullptr, nullptr, nullptr, lam);
        // ap = t * conj(W) + lam * p
        cgemm_wmma_kernel<<<gemmGrid, gemmBlk, 0, stream>>>(
            tr, ti, (long)HWSZ, Wr, Wi, 0L, apr, api, 0, 1,
            nullptr, pr, pi, lam);

        dot_kernel<<<NSLICE, 256, 0, stream>>>(pr, pi, apr, api, pAp);
        update_xr_kernel<<<elemGrid, elemBlk, 0, stream>>>(xr, xi, rr, ri, pr, pi, apr, api,
                                                           rTr, pAp, done);
        dot_kernel<<<NSLICE, 256, 0, stream>>>(rr, ri, rr, ri, rTrN);
        update_p_kernel<<<elemGrid, elemBlk, 0, stream>>>(pr, pi, rr, ri, rTrN, rTr, done);
        commit_rtr_kernel<<<1, 64, 0, stream>>>(rTr, rTrN, done);
    }

    writeback_kernel<<<elemGrid, elemBlk, 0, stream>>>(xr, xi, out);
}